// WGCNLayer_24635932410312
// MI455X (gfx1250) — compile-verified
//
#include <hip/hip_runtime.h>
#include <hip/hip_bf16.h>

typedef __attribute__((ext_vector_type(2))) float v2f;
typedef __attribute__((ext_vector_type(8))) float v8f;

#define D 128            // D_IN == D_OUT == 128
#define BN_EPS 1e-5f

// ---------------------------------------------------------------------------
// Kernel 0: zero agg (d_out) and stats (256 floats in ws)
// ---------------------------------------------------------------------------
__global__ __launch_bounds__(256) void k_zero(float4* __restrict__ out, size_t n4,
                                              float4* __restrict__ stats4) {
    size_t i = (size_t)blockIdx.x * blockDim.x + threadIdx.x;
    float4 z = make_float4(0.f, 0.f, 0.f, 0.f);
    if (i < n4) out[i] = z;
    if (i < 64) stats4[i] = z;   // 256 floats of stats
}

// ---------------------------------------------------------------------------
// Kernel 1: feats = x @ W  via V_WMMA_F32_16X16X4_F32
// block = 256 threads = 8 waves; block b -> M-tile b (16 rows), wave w -> N-tile w
// ---------------------------------------------------------------------------
__global__ __launch_bounds__(256) void k_gemm(const float* __restrict__ x,
                                              const float* __restrict__ w,
                                              float* __restrict__ feats) {
    const int lane = threadIdx.x & 31;
    const int wave = threadIdx.x >> 5;
    const int m0   = blockIdx.x << 4;     // 16-row band
    const int n0   = wave << 4;           // 16-col band (8 waves cover N=128)

    const int halfSel = lane >> 4;        // 0: lanes 0-15, 1: lanes 16-31
    const int lid     = lane & 15;
    const int kb      = halfSel << 1;     // k sub-offset 0 or 2

    const float* arow = x + (size_t)(m0 + lid) * D;   // A: row = m0+lid
    const int    bcol = n0 + lid;                      // B: col = n0+lid

    v8f c = {};
    #pragma unroll
    for (int k0 = 0; k0 < D; k0 += 4) {
        // A fragment: A[row][k0+kb], A[row][k0+kb+1]  (8B-aligned pair)
        v2f a = *(const v2f*)(arow + k0 + kb);
        // B fragment: W[k0+kb][col], W[k0+kb+1][col]
        v2f b;
        b.x = w[(size_t)(k0 + kb)     * D + bcol];
        b.y = w[(size_t)(k0 + kb + 1) * D + bcol];
        c = __builtin_amdgcn_wmma_f32_16x16x4_f32(
                /*neg_a=*/false, a, /*neg_b=*/false, b,
                /*c_mod=*/(short)0, c, /*reuse_a=*/false, /*reuse_b=*/false);
    }

    // C/D layout: VGPR r -> row (halfSel*8 + r), col = n0 + lid
    const int rbase = halfSel << 3;
    #pragma unroll
    for (int r = 0; r < 8; ++r)
        feats[(size_t)(m0 + rbase + r) * D + bcol] = c[r];
}

// ---------------------------------------------------------------------------
// Kernel 2: one wave per edge: agg[dst] += alp * feats[src]
// lane handles 4 consecutive channels (float4 gather, 4 hw f32 atomics)
// ---------------------------------------------------------------------------
__global__ __launch_bounds__(256) void k_edges(const float* __restrict__ feats,
                                               const float* __restrict__ alpha,
                                               const int* __restrict__ src,
                                               const int* __restrict__ dst,
                                               const int* __restrict__ etype,
                                               float* __restrict__ agg,
                                               int n_edges, int half_rel) {
    const int e    = (int)((blockIdx.x * (size_t)blockDim.x + threadIdx.x) >> 5);
    const int lane = threadIdx.x & 31;
    if (e >= n_edges) return;

    const int s = src[e];
    const int d = dst[e];
    const int t = etype[e];
    const int tt = (t >= half_rel) ? (t - half_rel) : (t + half_rel);
    const float alp = alpha[t] + alpha[tt];

    const float4 v = ((const float4*)(feats + (size_t)s * D))[lane];
    float* o = agg + (size_t)d * D + lane * 4;
    unsafeAtomicAdd(o + 0, v.x * alp);
    unsafeAtomicAdd(o + 1, v.y * alp);
    unsafeAtomicAdd(o + 2, v.z * alp);
    unsafeAtomicAdd(o + 3, v.w * alp);
}

// ---------------------------------------------------------------------------
// Kernel 3: per-column sum & sum-of-squares (biased batch stats)
// thread -> column (tid&127); 2 rows per block-iteration; partials via atomics
// ---------------------------------------------------------------------------
__global__ __launch_bounds__(256) void k_stats(const float* __restrict__ agg,
                                               float* __restrict__ stats,
                                               int n_nodes) {
    const int col       = threadIdx.x & (D - 1);
    const int rowsPerBlk = blockDim.x >> 7;              // 2
    int r          = blockIdx.x * rowsPerBlk + (threadIdx.x >> 7);
    const int step = gridDim.x * rowsPerBlk;

    float s = 0.f, ss = 0.f;
    for (; r < n_nodes; r += step) {
        float v = agg[(size_t)r * D + col];
        s += v; ss += v * v;
    }
    unsafeAtomicAdd(&stats[col],     s);
    unsafeAtomicAdd(&stats[D + col], ss);
}

// ---------------------------------------------------------------------------
// Kernel 4: in-place normalize: (v-mean)*rsqrt(var+eps)*gamma + beta
// ---------------------------------------------------------------------------
__global__ __launch_bounds__(256) void k_norm(float* __restrict__ agg,
                                              const float* __restrict__ stats,
                                              const float* __restrict__ gamma,
                                              const float* __restrict__ beta,
                                              int n_nodes) {
    const size_t n4 = (size_t)n_nodes * (D / 4);
    const size_t i  = (size_t)blockIdx.x * blockDim.x + threadIdx.x;
    if (i >= n4) return;

    const float invN = 1.0f / (float)n_nodes;
    const int colBase = ((int)(i & (D / 4 - 1))) << 2;

    float4 v = ((const float4*)agg)[i];
    float r[4] = {v.x, v.y, v.z, v.w};
    #pragma unroll
    for (int c = 0; c < 4; ++c) {
        const int col  = colBase + c;
        const float mu = stats[col] * invN;
        const float var = stats[D + col] * invN - mu * mu;
        const float sc  = gamma[col] * rsqrtf(var + BN_EPS);
        r[c] = (r[c] - mu) * sc + beta[col];
    }
    ((float4*)agg)[i] = make_float4(r[0], r[1], r[2], r[3]);
}

// ---------------------------------------------------------------------------
extern "C" void kernel_launch(void* const* d_in, const int* in_sizes, int n_in,
                              void* d_out, int out_size, void* d_ws, size_t ws_size,
                              hipStream_t stream) {
    const float* x      = (const float*)d_in[0];
    const float* weight = (const float*)d_in[1];
    const float* alpha  = (const float*)d_in[2];
    const float* gamma  = (const float*)d_in[3];
    const float* beta   = (const float*)d_in[4];
    const int*   src    = (const int*)d_in[5];
    const int*   dst    = (const int*)d_in[6];
    const int*   etype  = (const int*)d_in[7];

    const int n_nodes  = in_sizes[0] / D;       // 100000
    const int n_edges  = in_sizes[5];           // 1600000
    const int half_rel = in_sizes[2] / 2;       // 24

    float* agg   = (float*)d_out;                         // [n_nodes, 128]
    float* feats = (float*)d_ws;                          // [n_nodes, 128]
    float* stats = feats + (size_t)n_nodes * D;           // [256]

    // 0) zero agg + stats
    {
        size_t n4 = (size_t)n_nodes * (D / 4);
        int blocks = (int)((n4 + 255) / 256);
        k_zero<<<blocks, 256, 0, stream>>>((float4*)agg, n4, (float4*)stats);
    }
    // 1) feats = x @ W   (WMMA f32 16x16x4)
    k_gemm<<<n_nodes / 16, 256, 0, stream>>>(x, weight, feats);
    // 2) edge scatter (wave per edge)
    {
        int waves_per_block = 256 / 32;
        int blocks = (n_edges + waves_per_block - 1) / waves_per_block;
        k_edges<<<blocks, 256, 0, stream>>>(feats, alpha, src, dst, etype,
                                            agg, n_edges, half_rel);
    }
    // 3) column stats
    k_stats<<<512, 256, 0, stream>>>(agg, stats, n_nodes);
    // 4) normalize in place
    {
        size_t n4 = (size_t)n_nodes * (D / 4);
        int blocks = (int)((n4 + 255) / 256);
        k_norm<<<blocks, 256, 0, stream>>>(agg, stats, gamma, beta, n_nodes);
    }
}